// RNNDetector_86122684220301
// MI455X (gfx1250) — compile-verified
//
#include <hip/hip_runtime.h>

// RNNDetector: 2-layer tanh RNN (H=16, T=2^20) + linear head.
// Serial-latency-optimized: 1 workgroup, 3 wave32 waves pipelined by chunks.
//   wave0: layer-0 scan  (SGPR-broadcast state, split FMA chains, v_tanh)
//   wave1: xp1 = h1 @ W_ih1^T + b via V_WMMA_F32_16X16X4_F32, then layer-1 scan
//   wave2: logits head out = h2 @ W_lin^T + b_lin

#define T_LEN   1048576
#define H       16
#define CHUNK   256
#define NCHUNK  (T_LEN / CHUNK)

typedef __attribute__((ext_vector_type(2))) float v2f;
typedef __attribute__((ext_vector_type(8))) float v8f;

#if __has_builtin(__builtin_amdgcn_tanhf)
#define TANHF(x) __builtin_amdgcn_tanhf(x)
#else
#define TANHF(x) tanhf(x)
#endif

__device__ __forceinline__ float readlane_f(float v, int lane) {
  return __builtin_bit_cast(float,
      __builtin_amdgcn_readlane(__builtin_bit_cast(int, v), lane));
}

__global__ __launch_bounds__(96, 1)
void rnn_pipeline_kernel(const float* __restrict__ y,
                         const float* __restrict__ W_ih0, const float* __restrict__ b_ih0,
                         const float* __restrict__ W_hh0, const float* __restrict__ b_hh0,
                         const float* __restrict__ W_ih1, const float* __restrict__ b_ih1,
                         const float* __restrict__ W_hh1, const float* __restrict__ b_hh1,
                         const float* __restrict__ W_lin, const float* __restrict__ b_lin,
                         float* __restrict__ out)
{
  __shared__ float h1buf[2][CHUNK * H];   // wave0 -> wave1 (double buffered)
  __shared__ float h2buf[2][CHUNK * H];   // wave1 -> wave2 (double buffered)
  __shared__ float xp1buf[CHUNK * H];     // wave1-private per-chunk scratch

  const int tid  = threadIdx.x;
  const int wave = tid >> 5;      // wave32
  const int l    = tid & 31;
  const int n    = l & 15;        // logical row/col 0..15 (lanes 16-31 mirror)
  const int half = l >> 4;        // 0 = lanes 0-15, 1 = lanes 16-31

  // ---- per-wave preloaded constants (registers) ----
  float w0[H];   // wave0: W_hh0 row n
  float w1[H];   // wave1: W_hh1 row n
  float wl[H];   // wave2: W_lin row (l&1)
  v2f   Bw[4];   // wave1: B operand blocks for WMMA (W_ih1^T)
  float wih = 0.f, bs0 = 0.f, b1v = 0.f, bl = 0.f;

  float sh[H];   // broadcast hidden state (kept uniform via v_readlane -> SGPRs)
  #pragma unroll
  for (int k = 0; k < H; ++k) sh[k] = 0.0f;

  if (wave == 0) {
    #pragma unroll
    for (int k = 0; k < H; ++k) w0[k] = W_hh0[n * H + k];
    wih = W_ih0[n];                       // W_ih0 is [16,1]
    bs0 = b_ih0[n] + b_hh0[n];
  } else if (wave == 1) {
    #pragma unroll
    for (int k = 0; k < H; ++k) w1[k] = W_hh1[n * H + k];
    // B[k][nn] = W_ih1[nn][k]; lane l supplies N=n, K = 4*kb + 2*half + {0,1}
    #pragma unroll
    for (int kb = 0; kb < 4; ++kb) {
      const int K = 4 * kb + 2 * half;
      Bw[kb] = *(const v2f*)(W_ih1 + n * H + K);   // K contiguous in memory
    }
    b1v = b_ih1[n] + b_hh1[n];
  } else {
    const int o = l & 1;
    #pragma unroll
    for (int k = 0; k < H; ++k) wl[k] = W_lin[o * H + k];
    bl = b_lin[o];
  }

  for (int i = 0; i < NCHUNK + 2; ++i) {
    if (wave == 0) {
      // ---------------- layer-0 serial scan: chunk i ----------------
      if (i < NCHUNK) {
        float* hb = h1buf[i & 1];
        const int base = i * CHUNK;
        if (i + 1 < NCHUNK) __builtin_prefetch(y + base + CHUNK, 0, 0);
        for (int s = 0; s < CHUNK; ++s) {
          const float yv = y[base + s];             // uniform -> s_load
          float a0 = fmaf(yv, wih, bs0);            // xp0_t for row n
          float a1 = 0.f;
          #pragma unroll
          for (int k = 0; k < H; k += 2) {          // two 8-deep FMA chains
            a0 = fmaf(sh[k],     w0[k],     a0);
            a1 = fmaf(sh[k + 1], w0[k + 1], a1);
          }
          const float hn = TANHF(a0 + a1);
          if (l < 16) hb[s * H + n] = hn;
          #pragma unroll
          for (int k = 0; k < H; ++k) sh[k] = readlane_f(hn, k);
        }
      }
    } else if (wave == 1) {
      // -------- xp1 GEMM (WMMA) + layer-1 serial scan: chunk i-1 --------
      if (i >= 1 && i <= NCHUNK) {
        const int c = i - 1;
        const float* hb = h1buf[c & 1];
        float*       ob = h2buf[c & 1];

        // xp1 = h1_tile(16x16) @ W_ih1^T(16x16) + bias, K split into 4x K=4
        for (int tile = 0; tile < CHUNK / 16; ++tile) {
          v8f acc;
          #pragma unroll
          for (int v = 0; v < 8; ++v) acc[v] = b1v;  // C = bias (per column n)
          #pragma unroll
          for (int kb = 0; kb < 4; ++kb) {
            const int K = 4 * kb + 2 * half;
            // A 16x4 f32 layout: lane n holds (K, K+1) of row (tile*16+n)
            const v2f Aa = *(const v2f*)(hb + (tile * 16 + n) * H + K);
            acc = __builtin_amdgcn_wmma_f32_16x16x4_f32(
                false, Aa, false, Bw[kb], (short)0, acc, false, false);
          }
          // D layout: VGPR v -> row v + 8*half, col n
          #pragma unroll
          for (int v = 0; v < 8; ++v)
            xp1buf[(tile * 16 + v + 8 * half) * H + n] = acc[v];
        }

        // serial scan over the chunk (same-wave LDS RAW is in-order)
        for (int s = 0; s < CHUNK; ++s) {
          float a0 = xp1buf[s * H + n];
          float a1 = 0.f;
          #pragma unroll
          for (int k = 0; k < H; k += 2) {
            a0 = fmaf(sh[k],     w1[k],     a0);
            a1 = fmaf(sh[k + 1], w1[k + 1], a1);
          }
          const float hn = TANHF(a0 + a1);
          if (l < 16) ob[s * H + n] = hn;
          #pragma unroll
          for (int k = 0; k < H; ++k) sh[k] = readlane_f(hn, k);
        }
      }
    } else {
      // ---------------- linear head: chunk i-2 ----------------
      if (i >= 2) {
        const int c = i - 2;
        const float* hb = h2buf[c & 1];
        const int o  = l & 1;
        const int tt = l >> 1;
        const int base = c * CHUNK;
        for (int s = tt; s < CHUNK; s += 16) {
          float acc = bl;
          #pragma unroll
          for (int k = 0; k < H; ++k)
            acc = fmaf(hb[s * H + k], wl[k], acc);
          out[(base + s) * 2 + o] = acc;
        }
      }
    }
    __syncthreads();
  }
}

extern "C" void kernel_launch(void* const* d_in, const int* in_sizes, int n_in,
                              void* d_out, int out_size, void* d_ws, size_t ws_size,
                              hipStream_t stream) {
  (void)in_sizes; (void)n_in; (void)out_size; (void)d_ws; (void)ws_size;
  const float* y     = (const float*)d_in[0];
  const float* W_ih0 = (const float*)d_in[1];
  const float* b_ih0 = (const float*)d_in[2];
  const float* W_hh0 = (const float*)d_in[3];
  const float* b_hh0 = (const float*)d_in[4];
  const float* W_ih1 = (const float*)d_in[5];
  const float* b_ih1 = (const float*)d_in[6];
  const float* W_hh1 = (const float*)d_in[7];
  const float* b_hh1 = (const float*)d_in[8];
  const float* W_lin = (const float*)d_in[9];
  const float* b_lin = (const float*)d_in[10];
  float* out = (float*)d_out;

  rnn_pipeline_kernel<<<dim3(1), dim3(96), 0, stream>>>(
      y, W_ih0, b_ih0, W_hh0, b_hh0, W_ih1, b_ih1, W_hh1, b_hh1,
      W_lin, b_lin, out);
}